// PMTGCN_3135326126732
// MI455X (gfx1250) — compile-verified
//
#include <hip/hip_runtime.h>
#include <math.h>

// ---------------------------------------------------------------------------
// PMT-GCN forward for MI455X (gfx1250, wave32, WMMA f32 16x16x4)
// ---------------------------------------------------------------------------

typedef float v2f __attribute__((ext_vector_type(2)));
typedef float v8f __attribute__((ext_vector_type(8)));

__device__ __forceinline__ void atomicAddF(float* p, float v) {
  __hip_atomic_fetch_add(p, v, __ATOMIC_RELAXED, __HIP_MEMORY_SCOPE_AGENT);
}

__device__ __forceinline__ float gelu_exact(float v) {
  // matches jax.nn.gelu(approximate=False): 0.5*x*(1+erf(x/sqrt(2)))
  return 0.5f * v * (1.0f + erff(v * 0.70710678118654752f));
}

// ------------------------- fused GEMM: Y = epi(X' @ W + b) ------------------
// X: [nrows,128] row-major; optional per-channel affine on A:  X' = X*as + at
// W: [128,NCOLS] row-major; bias: [NCOLS] or null
// residual: [nrows,NCOLS] or null, with optional per-channel affine (rs, rt),
//           added AFTER the optional GELU.
// W is staged into LDS in K-pair-major swizzled form so each lane's B fragment
// {W[k][c], W[k+1][c]} is one aligned ds_load_b64 (bank-conflict-free: the
// +32-dword row pad puts odd K-pair rows in banks 32..63).
// FULL=true  -> all 128 rows of this block are valid (no guards anywhere)
// grid.x full blocks; block = 256 threads (8 waves, 16-row stripe per wave).
template <int NCOLS, bool FULL>
__global__ __launch_bounds__(256) void gemm_fused(
    const float* __restrict__ X, const float* __restrict__ W,
    const float* __restrict__ bias,
    const float* __restrict__ ascale, const float* __restrict__ ashift,
    const float* __restrict__ residual,
    const float* __restrict__ rscale, const float* __restrict__ rshift,
    float* __restrict__ Y, int nrows, int act, int blockOff) {
  constexpr int SWS = 2 * NCOLS + 32;  // dwords per K-pair row (pad = 32)
  __shared__ float sW[64 * SWS];       // K=128 -> 64 K-pair rows

  const int tid = threadIdx.x;
  // stage + swizzle: element (k, c) -> sW[(k>>1)*SWS + 2*c + (k&1)]
  for (int i = tid * 4; i < 128 * NCOLS; i += 256 * 4) {
    const float4 w = *(const float4*)&W[i];
    const int k = i / NCOLS;
    const int c = i - k * NCOLS;
    float* p = &sW[(k >> 1) * SWS + 2 * c + (k & 1)];
    p[0] = w.x; p[2] = w.y; p[4] = w.z; p[6] = w.w;
  }
  __syncthreads();

  const int wave = tid >> 5;
  const int lane = tid & 31;
  const int lh = lane >> 4;   // half-wave selector
  const int ll = lane & 15;
  const int rowBase = (blockIdx.x + blockOff) * 128 + wave * 16;
  const int arow = rowBase + ll;
  const bool rowok = FULL ? true : (arow < nrows);

  // A fragments: ISA 16x4 fp32 layout -> lanes 0-15: K=4k+{0,1}; 16-31: K=4k+{2,3}
  v2f afrag[32];
  const float* xr = X + (size_t)(rowok ? arow : 0) * 128 + 2 * lh;
#pragma unroll
  for (int ks = 0; ks < 32; ++ks) {
    v2f a;
    a.x = xr[4 * ks + 0];
    a.y = xr[4 * ks + 1];
    if (!FULL && !rowok) { a.x = 0.0f; a.y = 0.0f; }
    if (ascale) {  // fused BatchNorm on the A operand (per-channel = per-K)
      const int k0 = 4 * ks + 2 * lh;
      a.x = fmaf(a.x, ascale[k0], ashift[k0]);
      a.y = fmaf(a.y, ascale[k0 + 1], ashift[k0 + 1]);
    }
    afrag[ks] = a;
  }

  // four concurrent column tiles -> four independent WMMA accumulation chains
  for (int ct = 0; ct < NCOLS / 16; ct += 4) {
    int col[4];
    v8f acc[4];
#pragma unroll
    for (int j = 0; j < 4; ++j) {
      col[j] = (ct + j) * 16 + ll;
      acc[j] = {};
    }
#pragma unroll
    for (int ks = 0; ks < 32; ++ks) {
      // B fragment for K-step ks: rows kr=4ks+2lh, kr+1 at column col[j]
      // -> swizzled K-pair row (2ks+lh), one aligned b64 per lane.
      const int kp = 2 * ks + lh;
#pragma unroll
      for (int j = 0; j < 4; ++j) {
        const v2f b = *(const v2f*)&sW[kp * SWS + 2 * col[j]];
        acc[j] = __builtin_amdgcn_wmma_f32_16x16x4_f32(
            false, afrag[ks], false, b, (short)0, acc[j], false, false);
      }
    }
    // epilogue: C/D layout -> VGPR r holds rows (r, r+8); lane&15 selects col
#pragma unroll
    for (int j = 0; j < 4; ++j) {
      const int c = col[j];
      const float bc = bias ? bias[c] : 0.0f;
      const float rs = rscale ? rscale[c] : 1.0f;
      const float rt = rshift ? rshift[c] : 0.0f;
#pragma unroll
      for (int r = 0; r < 8; ++r) {
        const int row = rowBase + r + 8 * lh;
        if (FULL || row < nrows) {
          float v = acc[j][r] + bc;
          if (act) v = gelu_exact(v);
          if (residual) {
            float rv = residual[(size_t)row * NCOLS + c];
            v += rscale ? fmaf(rv, rs, rt) : rv;
          }
          Y[(size_t)row * NCOLS + c] = v;
        }
      }
    }
  }
}

template <int NCOLS>
static void launch_gemm(const float* X, const float* W, const float* bias,
                        const float* ascale, const float* ashift,
                        const float* residual, const float* rscale,
                        const float* rshift, float* Y, int nrows, int act,
                        hipStream_t stream) {
  const int nFull = nrows / 128;
  if (nFull > 0) {
    gemm_fused<NCOLS, true><<<nFull, 256, 0, stream>>>(
        X, W, bias, ascale, ashift, residual, rscale, rshift, Y, nrows, act, 0);
  }
  if (nrows - nFull * 128 > 0) {
    gemm_fused<NCOLS, false><<<1, 256, 0, stream>>>(
        X, W, bias, ascale, ashift, residual, rscale, rshift, Y, nrows, act,
        nFull);
  }
}

// ------------------------------ small kernels -------------------------------

__global__ void fill_f32(float* __restrict__ p, float v, int n) {
  int i = blockIdx.x * blockDim.x + threadIdx.x;
  if (i < n) p[i] = v;
}

__global__ void deg_acc(const int* __restrict__ dst, float* __restrict__ deg,
                        int ne) {
  int e = blockIdx.x * blockDim.x + threadIdx.x;
  if (e < ne) atomicAddF(&deg[dst[e]], 1.0f);
}

__global__ void rsqrt_inplace(float* __restrict__ p, int n) {
  int i = blockIdx.x * blockDim.x + threadIdx.x;
  if (i < n) p[i] = rsqrtf(p[i]);  // deg >= 1 always (self-loop)
}

// one edge per 32-thread group, 4 contiguous floats per thread (float4 gather)
__global__ void scatter_edges(const float* __restrict__ xw,
                              const int* __restrict__ src,
                              const int* __restrict__ dst,
                              const float* __restrict__ dis,
                              float* __restrict__ agg, int ne) {
  int t = blockIdx.x * blockDim.x + threadIdx.x;
  int e = t >> 5, lane = t & 31;
  if (e >= ne) return;
  int s = src[e], d = dst[e];
  float nrm = dis[s] * dis[d];
  const float4 v = *(const float4*)(xw + (size_t)s * 128 + lane * 4);
  float* p = agg + (size_t)d * 128 + lane * 4;
  atomicAddF(p + 0, v.x * nrm);
  atomicAddF(p + 1, v.y * nrm);
  atomicAddF(p + 2, v.z * nrm);
  atomicAddF(p + 3, v.w * nrm);
}

// x = agg + xw*dis^2 (self-loop term) + conv_b + x   (residual, in place)
__global__ void conv_combine(const float* __restrict__ agg,
                             const float* __restrict__ xw,
                             const float* __restrict__ dis,
                             const float* __restrict__ cb,
                             float* __restrict__ x, int total) {
  int i = blockIdx.x * blockDim.x + threadIdx.x;
  if (i < total) {
    int node = i >> 7, c = i & 127;
    float dn = dis[node];
    x[i] = agg[i] + xw[i] * dn * dn + cb[c] + x[i];
  }
}

// 128 threads/block (one per channel), each block reduces 256 rows
__global__ void bn_partial(const float* __restrict__ x, float* __restrict__ sums,
                           float* __restrict__ sqs, int nrows) {
  int c = threadIdx.x;
  int r0 = blockIdx.x * 256;
  int r1 = r0 + 256; if (r1 > nrows) r1 = nrows;
  float s = 0.0f, q = 0.0f;
  for (int r = r0; r < r1; ++r) {
    float v = x[(size_t)r * 128 + c];
    s += v; q += v * v;
  }
  atomicAddF(&sums[c], s);
  atomicAddF(&sqs[c], q);
}

__global__ void bn_finalize(const float* __restrict__ sums,
                            const float* __restrict__ sqs,
                            const float* __restrict__ gamma,
                            const float* __restrict__ beta,
                            float* __restrict__ scale,
                            float* __restrict__ shift, float invN) {
  int c = threadIdx.x;
  float mean = sums[c] * invN;
  float var = sqs[c] * invN - mean * mean;  // biased variance (ddof=0)
  float inv = rsqrtf(var + 1e-5f);
  float sc = gamma[c] * inv;
  scale[c] = sc;
  shift[c] = beta[c] - mean * sc;
}

// one node per 32-thread group
__global__ void pool_acc(const float* __restrict__ x,
                         const int* __restrict__ batch, float* __restrict__ ps,
                         float* __restrict__ cnt, int n) {
  int t = blockIdx.x * blockDim.x + threadIdx.x;
  int node = t >> 5, lane = t & 31;
  if (node >= n) return;
  int g = batch[node];
  const float4 v = *(const float4*)(x + (size_t)node * 128 + lane * 4);
  float* p = ps + (size_t)g * 128 + lane * 4;
  atomicAddF(p + 0, v.x);
  atomicAddF(p + 1, v.y);
  atomicAddF(p + 2, v.z);
  atomicAddF(p + 3, v.w);
  if (lane == 0) atomicAddF(&cnt[g], 1.0f);
}

__global__ void pool_div(const float* __restrict__ ps,
                         const float* __restrict__ cnt,
                         float* __restrict__ pooled, int total) {
  int i = blockIdx.x * blockDim.x + threadIdx.x;
  if (i < total) {
    int g = i >> 7;
    pooled[i] = ps[i] / fmaxf(cnt[g], 1.0f);
  }
}

// ---------------------------------------------------------------------------

static inline int cdiv(int a, int b) { return (a + b - 1) / b; }

extern "C" void kernel_launch(void* const* d_in, const int* in_sizes, int n_in,
                              void* d_out, int out_size, void* d_ws,
                              size_t ws_size, hipStream_t stream) {
  // inputs (setup_inputs order)
  const float* x_in    = (const float*)d_in[0];
  const int*   ei      = (const int*)d_in[1];
  const int*   batch   = (const int*)d_in[2];
  const float* pre_w1  = (const float*)d_in[3];
  const float* pre_b1  = (const float*)d_in[4];
  const float* pre_w2  = (const float*)d_in[5];
  const float* pre_b2  = (const float*)d_in[6];
  const float* conv_w  = (const float*)d_in[7];
  const float* conv_b  = (const float*)d_in[8];
  const float* ffnn_w1 = (const float*)d_in[9];
  const float* ffnn_b1 = (const float*)d_in[10];
  const float* ffnn_w2 = (const float*)d_in[11];
  const float* ffnn_b2 = (const float*)d_in[12];
  const float* bn_g    = (const float*)d_in[13];
  const float* bn_b    = (const float*)d_in[14];
  const float* post_w1 = (const float*)d_in[15];
  const float* post_b1 = (const float*)d_in[16];
  const float* post_w2 = (const float*)d_in[17];
  const float* post_b2 = (const float*)d_in[18];

  const int N    = in_sizes[0] / 128;
  const int E    = in_sizes[1] / 2;
  const int HOPS = in_sizes[8] / 128;   // conv_b is (HOPS,128)
  const int G    = out_size / 64;       // output (G, 64)
  const int NF   = N * 128;

  const int* esrc = ei;
  const int* edst = ei + E;

  // carve workspace
  char* ws = (char*)d_ws;
  size_t off = 0;
  auto carve = [&](size_t bytes) -> float* {
    float* p = (float*)(ws + off);
    off += (bytes + 255) & ~(size_t)255;
    return p;
  };
  float* bufA   = carve((size_t)NF * 4);
  float* bufB   = carve((size_t)NF * 4);
  float* bufC   = carve((size_t)NF * 4);
  float* dis    = carve((size_t)N * 4);
  float* bnsum  = carve(128 * 4);
  float* bnsq   = carve(128 * 4);
  float* bnsc   = carve(128 * 4);
  float* bnsh   = carve(128 * 4);
  float* ps     = carve((size_t)G * 128 * 4);
  float* cnt    = carve((size_t)G * 4);
  float* pooled = carve((size_t)G * 128 * 4);
  float* ph     = carve((size_t)G * 128 * 4);
  (void)ws_size; (void)n_in;

  // ---- degree / symmetric normalization: dis[n] = rsqrt(1 + in-degree) ----
  fill_f32<<<cdiv(N, 256), 256, 0, stream>>>(dis, 1.0f, N);  // self-loops
  deg_acc<<<cdiv(E, 256), 256, 0, stream>>>(edst, dis, E);
  rsqrt_inplace<<<cdiv(N, 256), 256, 0, stream>>>(dis, N);

  // ---- preprocess FFNN ----
  launch_gemm<128>(x_in, pre_w1, pre_b1, nullptr, nullptr, nullptr, nullptr,
                   nullptr, bufB, N, 1, stream);
  launch_gemm<128>(bufB, pre_w2, pre_b2, nullptr, nullptr, nullptr, nullptr,
                   nullptr, bufA, N, 0, stream);

  float* X = bufA;  // current features (pre-BN inside hop body)
  float* T = bufB;  // xw / ffnn hidden
  float* S = bufC;  // agg / ffnn output

  for (int h = 0; h < HOPS; ++h) {
    const float* cw  = conv_w + (size_t)h * 128 * 128;
    const float* cb  = conv_b + (size_t)h * 128;
    const float* fw1 = ffnn_w1 + (size_t)h * 128 * 128;
    const float* fb1 = ffnn_b1 + (size_t)h * 128;
    const float* fw2 = ffnn_w2 + (size_t)h * 128 * 128;
    const float* fb2 = ffnn_b2 + (size_t)h * 128;

    // GCNConv: xw = X @ W ; scatter-add over edges ; + self-loop + bias + res
    launch_gemm<128>(X, cw, nullptr, nullptr, nullptr, nullptr, nullptr,
                     nullptr, T, N, 0, stream);
    fill_f32<<<cdiv(NF, 256), 256, 0, stream>>>(S, 0.0f, NF);
    scatter_edges<<<cdiv(E * 32, 256), 256, 0, stream>>>(T, esrc, edst, dis, S, E);
    conv_combine<<<cdiv(NF, 256), 256, 0, stream>>>(S, T, dis, cb, X, NF);

    // BatchNorm statistics of post-residual X (pre-BN tensor)
    fill_f32<<<1, 256, 0, stream>>>(bnsum, 0.0f, 128);
    fill_f32<<<1, 256, 0, stream>>>(bnsq, 0.0f, 128);
    bn_partial<<<cdiv(N, 256), 128, 0, stream>>>(X, bnsum, bnsq, N);
    bn_finalize<<<1, 128, 0, stream>>>(bnsum, bnsq, bn_g + (size_t)h * 128,
                                       bn_b + (size_t)h * 128, bnsc, bnsh,
                                       1.0f / (float)N);

    // FFNN with residual; BN folded into A-operand of gemm1 and the residual
    // read of gemm2 (X stays pre-BN, never materializing BN(x)).
    launch_gemm<128>(X, fw1, fb1, bnsc, bnsh, nullptr, nullptr, nullptr, T, N,
                     1, stream);
    launch_gemm<128>(T, fw2, fb2, nullptr, nullptr, X, bnsc, bnsh, S, N, 0,
                     stream);
    float* tmp = X; X = S; S = tmp;
  }

  // ---- global mean pool ----
  fill_f32<<<cdiv(G * 128, 256), 256, 0, stream>>>(ps, 0.0f, G * 128);
  fill_f32<<<1, 256, 0, stream>>>(cnt, 0.0f, G);
  pool_acc<<<cdiv(N * 32, 256), 256, 0, stream>>>(X, batch, ps, cnt, N);
  pool_div<<<cdiv(G * 128, 256), 256, 0, stream>>>(ps, cnt, pooled, G * 128);

  // ---- postprocess FFNN -> d_out [G,64] ----
  launch_gemm<128>(pooled, post_w1, post_b1, nullptr, nullptr, nullptr, nullptr,
                   nullptr, ph, G, 1, stream);
  launch_gemm<64>(ph, post_w2, post_b2, nullptr, nullptr, nullptr, nullptr,
                  nullptr, (float*)d_out, G, 0, stream);
}